// ModelWrapper_67242007986223
// MI455X (gfx1250) — compile-verified
//
#include <hip/hip_runtime.h>

#define N_UP     32768
#define N_LR     10475
#define N_JOINT  55
#define EXPR_DIM 50

typedef __attribute__((ext_vector_type(2))) float v2f;
typedef __attribute__((ext_vector_type(8))) float v8f;

// ---- output layout (floats), concatenated in reference return order ----
#define O_MEAN3D    0
#define O_OPACITY   (N_UP*3)
#define O_SCALE     (O_OPACITY + N_UP)
#define O_ROT       (O_SCALE + N_UP*3)
#define O_RGB       (O_ROT + N_UP*4)
#define O_MEANREF   (O_RGB + N_UP*3)
#define O_SCALEREF  (O_MEANREF + N_UP*3)
#define O_SKIN      (O_SCALEREF + N_UP*3)

// ======================= Kernel A: pointwise =======================
__global__ __launch_bounds__(256) void pointwise_kernel(
    const float* __restrict__ mean_base,
    const float* __restrict__ scale_log,
    const float* __restrict__ scale_off,
    const float* __restrict__ rgb_raw,
    const float* __restrict__ comb_off,
    const float* __restrict__ expr,
    const float* __restrict__ expr_dirs,
    float* __restrict__ out)
{
    int n = blockIdx.x * blockDim.x + threadIdx.x;
    if (n >= N_UP) return;

    // expr_off[d] = sum_e expr[e] * expr_dirs[n,d,e]
    const float* ed = expr_dirs + (size_t)n * (3 * EXPR_DIM);
    float e0 = 0.f, e1 = 0.f, e2 = 0.f;
#pragma unroll 5
    for (int e = 0; e < EXPR_DIM; ++e) {
        float x = expr[e];                    // uniform -> scalar loads
        e0 = fmaf(x, ed[e], e0);
        e1 = fmaf(x, ed[EXPR_DIM + e], e1);
        e2 = fmaf(x, ed[2 * EXPR_DIM + e], e2);
    }

    float m0 = mean_base[n*3+0] + e0;
    float m1 = mean_base[n*3+1] + e1;
    float m2 = mean_base[n*3+2] + e2;
    out[O_MEAN3D + n*3+0] = m0;
    out[O_MEAN3D + n*3+1] = m1;
    out[O_MEAN3D + n*3+2] = m2;

    out[O_MEANREF + n*3+0] = m0 + comb_off[n*3+0];
    out[O_MEANREF + n*3+1] = m1 + comb_off[n*3+1];
    out[O_MEANREF + n*3+2] = m2 + comb_off[n*3+2];

    float sl = scale_log[n];
    float s  = expf(sl);
    float sr = expf(sl + scale_off[n]);
    out[O_SCALE + n*3+0] = s;  out[O_SCALE + n*3+1] = s;  out[O_SCALE + n*3+2] = s;
    out[O_SCALEREF + n*3+0] = sr; out[O_SCALEREF + n*3+1] = sr; out[O_SCALEREF + n*3+2] = sr;

    out[O_RGB + n*3+0] = (tanhf(rgb_raw[n*3+0]) + 1.0f) * 0.5f;
    out[O_RGB + n*3+1] = (tanhf(rgb_raw[n*3+1]) + 1.0f) * 0.5f;
    out[O_RGB + n*3+2] = (tanhf(rgb_raw[n*3+2]) + 1.0f) * 0.5f;

    out[O_OPACITY + n] = 1.0f;
    out[O_ROT + n*4+0] = 1.0f;
    out[O_ROT + n*4+1] = 0.0f;
    out[O_ROT + n*4+2] = 0.0f;
    out[O_ROT + n*4+3] = 0.0f;
}

// ======================= Kernel B: WMMA NN + skin gather =======================
// Score tile via V_WMMA_F32_16X16X4_F32:
//   A (16x4): row m = [px,py,pz,1]    -> lane<16: {p.x,p.y}(K0,K1); lane>=16: {p.z,1}(K2,K3)
//   B (4x16): col j = [-2qx,-2qy,-2qz,|q|^2]
//   C[m][j]  = -2 p.q + |q|^2  (== d2 minus per-row constant |p|^2 -> same argmin)
__device__ inline v8f tile_scores(v2f a, v2f b, int lane) {
#if __has_builtin(__builtin_amdgcn_wmma_f32_16x16x4_f32)
    v8f z = {};
    return __builtin_amdgcn_wmma_f32_16x16x4_f32(false, a, false, b, (short)0, z,
                                                 false, false);
#else
    // shuffle-based emulation of the same lane layout
    v8f c;
    int n = lane & 15;
    float B0 = __shfl(a.x * 0.f + b.x, n, 32);  // keep types simple
    float B1 = __shfl(b.y, n, 32);
    float B2 = __shfl(b.x, n + 16, 32);
    float B3 = __shfl(b.y, n + 16, 32);
    int hi = lane >> 4;
#pragma unroll
    for (int r = 0; r < 8; ++r) {
        int m = r + hi * 8;
        float A0 = __shfl(a.x, m, 32);
        float A1 = __shfl(a.y, m, 32);
        float A2 = __shfl(a.x, m + 16, 32);
        float A3 = __shfl(a.y, m + 16, 32);
        c[r] = fmaf(A0, B0, fmaf(A1, B1, fmaf(A2, B2, A3 * B3)));
    }
    return c;
#endif
}

#define TILE_C 2048            // candidates per LDS tile (32 KB)
#define NUM_TILES 6            // 6*2048 = 12288 >= N_LR (padded with +inf norm)

__global__ __launch_bounds__(256) void nn_skin_kernel(
    const float* __restrict__ mean3d,          // = d_out + O_MEAN3D (written by kernel A)
    const float* __restrict__ mesh_lr,
    const int*   __restrict__ lr_to_hr,
    const unsigned char* __restrict__ mask,
    const float* __restrict__ skinW,
    float* __restrict__ out)
{
    __shared__ float tileB[TILE_C * 4];

    int tid  = threadIdx.x;
    int wave = tid >> 5;
    int lane = tid & 31;
    int hi16 = lane >> 4;
    int l15  = lane & 15;
    int base = (blockIdx.x * 8 + wave) * 16;   // 16 points per wave

    // A fragment for this wave's 16 points
    int mrow = base + l15;
    v2f a;
    if (hi16 == 0) { a.x = mean3d[mrow*3+0]; a.y = mean3d[mrow*3+1]; }
    else           { a.x = mean3d[mrow*3+2]; a.y = 1.0f; }

    float minv[8];
    int   mini[8];
#pragma unroll
    for (int r = 0; r < 8; ++r) { minv[r] = 3.0e38f; mini[r] = 0; }

    for (int t = 0; t < NUM_TILES; ++t) {
        // cooperative tile load + transform to B format
        for (int i = tid; i < TILE_C; i += 256) {
            int j = t * TILE_C + i;
            float qx = 0.f, qy = 0.f, qz = 0.f, nq = 1.0e30f;
            if (j < N_LR) {
                qx = mesh_lr[j*3+0]; qy = mesh_lr[j*3+1]; qz = mesh_lr[j*3+2];
                nq = qx*qx + qy*qy + qz*qz;
                qx *= -2.0f; qy *= -2.0f; qz *= -2.0f;
            }
            tileB[i*4+0] = qx; tileB[i*4+1] = qy;
            tileB[i*4+2] = qz; tileB[i*4+3] = nq;
        }
        __syncthreads();

#pragma unroll 4
        for (int c = 0; c < TILE_C / 16; ++c) {
            // B fragment: lane<16 -> {-2qx,-2qy} of col (c*16+l15); lane>=16 -> {-2qz,|q|^2}
            v2f b = *(const v2f*)&tileB[(c*16 + l15)*4 + hi16*2];
            v8f s = tile_scores(a, b, lane);
            int jcol = t * TILE_C + c*16 + l15;
#pragma unroll
            for (int r = 0; r < 8; ++r) {
                if (s[r] < minv[r]) { minv[r] = s[r]; mini[r] = jcol; }
            }
        }
        __syncthreads();
    }

    // argmin reduction across the 16 column-lanes of each half-wave
#pragma unroll
    for (int off = 1; off <= 8; off <<= 1) {
#pragma unroll
        for (int r = 0; r < 8; ++r) {
            float ov = __shfl_xor(minv[r], off, 32);
            int   oi = __shfl_xor(mini[r], off, 32);
            if (ov < minv[r] || (ov == minv[r] && oi < mini[r])) {
                minv[r] = ov; mini[r] = oi;
            }
        }
    }
    // lanes 0..15 now hold rows 0..7 (index r), lanes 16..31 hold rows 8..15

    // remap + skinning gather: wave copies 16 rows of 55 floats
#pragma unroll
    for (int m = 0; m < 16; ++m) {
        int bi = (m < 8) ? __shfl(mini[m], 0, 32) : __shfl(mini[m - 8], 16, 32);
        int n  = base + m;
        int nn = mask[n] ? n : lr_to_hr[bi];
        const float* src = skinW + (size_t)nn * N_JOINT;
        float*       dst = out + O_SKIN + (size_t)n * N_JOINT;
        for (int k = lane; k < N_JOINT; k += 32) dst[k] = src[k];
    }
}

// ======================= launch =======================
extern "C" void kernel_launch(void* const* d_in, const int* in_sizes, int n_in,
                              void* d_out, int out_size, void* d_ws, size_t ws_size,
                              hipStream_t stream) {
    const float* mean_base = (const float*)d_in[0];
    const float* mesh_lr   = (const float*)d_in[1];
    const float* scale_log = (const float*)d_in[2];
    const float* scale_off = (const float*)d_in[3];
    const float* rgb_raw   = (const float*)d_in[4];
    const float* comb_off  = (const float*)d_in[5];
    const float* expr      = (const float*)d_in[6];
    const float* expr_dirs = (const float*)d_in[7];
    const int*   lr_to_hr  = (const int*)d_in[8];
    const unsigned char* mask = (const unsigned char*)d_in[9];
    const float* skinW     = (const float*)d_in[10];
    float* out = (float*)d_out;

    (void)in_sizes; (void)n_in; (void)out_size; (void)d_ws; (void)ws_size;

    pointwise_kernel<<<N_UP / 256, 256, 0, stream>>>(
        mean_base, scale_log, scale_off, rgb_raw, comb_off, expr, expr_dirs, out);

    // 16 points/wave * 8 waves/block = 128 points per block
    nn_skin_kernel<<<N_UP / 128, 256, 0, stream>>>(
        out + O_MEAN3D, mesh_lr, lr_to_hr, mask, skinW, out);
}